// MultiHeadAttention_33913061769381
// MI455X (gfx1250) — compile-verified
//
#include <hip/hip_runtime.h>
#include <math.h>
#include <stdint.h>

// MI455X / gfx1250, wave32. Exact-fp32 MHA using V_WMMA_F32_16X16X4_F32
// plus CDNA5 async global->LDS staging (GLOBAL_LOAD_ASYNC_TO_LDS_B128).
// d_ws usage: 3 * B*S*D floats = 48 MB (qh | kh | vh in [b,h,s,hd] layout).

typedef __attribute__((ext_vector_type(2))) float v2f;
typedef __attribute__((ext_vector_type(8))) float v8f;

#define NH   16
#define NB   2
#define SEQ  2048
#define DIM  1024
#define HD   64
#define SCALE 0.125f   // 1/sqrt(64)

#define KSTRIDE 68     // K-tile LDS row stride (floats): conflict-free v2f reads
#define VSTRIDE 72     // V-tile LDS row stride (floats): conflict-free b32 reads
#define PSTRIDE 18     // P-tile LDS row stride (floats): conflict-free + 8B align

static __device__ __forceinline__ v8f wmma4(v2f a, v2f b, v8f c) {
  // D = A(16x4,f32) * B(4x16,f32) + C(16x16,f32)
  return __builtin_amdgcn_wmma_f32_16x16x4_f32(false, a, false, b, (short)0, c,
                                               false, false);
}

static __device__ __forceinline__ v8f v8f_zero() {
  v8f z = {0.f, 0.f, 0.f, 0.f, 0.f, 0.f, 0.f, 0.f};
  return z;
}

// Async copy of 16 bytes global->LDS, tracked by ASYNCcnt.
static __device__ __forceinline__ void async_b128(uint32_t lds_off,
                                                  const void* gaddr) {
  asm volatile("global_load_async_to_lds_b128 %0, %1, off"
               :
               : "v"(lds_off), "v"(gaddr)
               : "memory");
}

static __device__ __forceinline__ void wait_asynccnt_le2() {
  asm volatile("s_wait_asynccnt 0x2" ::: "memory");
}

// ---------------------------------------------------------------------------
// Fused QKV projection: out = x @ w.T, written as [b, h, s, hd].
// NT GEMM: out[m,n] = dot(x[m,:], w[n,:]).  A/B fragments for f32 WMMA are
// v2f loads at (row*D + k + 2*hiHalf) per the ISA 16x4 / 4x16 layouts.
// One wave computes a 16x64 strip (A fragment reused across 4 B tiles).
// grid = (32, 16, 3): 32 * 8 waves = 256 m-tiles, 16 n-groups of 64, z = q/k/v
// ---------------------------------------------------------------------------
__global__ __launch_bounds__(256) void proj_qkv_kernel(
    const float* __restrict__ q, const float* __restrict__ k,
    const float* __restrict__ v, const float* __restrict__ wq,
    const float* __restrict__ wk, const float* __restrict__ wv,
    float* __restrict__ qh, float* __restrict__ kh, float* __restrict__ vh) {
  const int lane = threadIdx.x & 31;
  const int wave = threadIdx.x >> 5;
  const int hi   = lane >> 4;   // which half of the wave (K-offset +2)
  const int lo   = lane & 15;   // row (A) / col (B) within the 16-tile

  const float* x; const float* w; float* o;
  switch (blockIdx.z) {
    case 0:  x = q; w = wq; o = qh; break;
    case 1:  x = k; w = wk; o = kh; break;
    default: x = v; w = wv; o = vh; break;
  }

  const int m0 = (blockIdx.x * 8 + wave) * 16;  // row-tile start in [0, NB*SEQ)
  const int n0 = blockIdx.y * 64;               // col-group start in [0, DIM)

  const float* arow = x + (size_t)(m0 + lo) * DIM + 2 * hi;
  const float* brow = w + (size_t)(n0 + lo) * DIM + 2 * hi;

  v8f acc[4] = {v8f_zero(), v8f_zero(), v8f_zero(), v8f_zero()};

#pragma unroll 4
  for (int kk = 0; kk < DIM; kk += 4) {
    v2f a = *(const v2f*)(arow + kk);
#pragma unroll
    for (int nt = 0; nt < 4; ++nt) {
      v2f b = *(const v2f*)(brow + (size_t)nt * 16 * DIM + kk);
      acc[nt] = wmma4(a, b, acc[nt]);
    }
  }

  // C/D layout: acc[r] @ lane -> row (r + 8*hi), col lo.  Store to [b,h,s,hd].
#pragma unroll
  for (int r = 0; r < 8; ++r) {
    const int gm = m0 + r + 8 * hi;
    const int bb = gm / SEQ;
    const int ss = gm % SEQ;
#pragma unroll
    for (int nt = 0; nt < 4; ++nt) {
      const int n  = n0 + nt * 16 + lo;
      const int h  = n >> 6;
      const int ch = n & 63;
      o[(((size_t)bb * NH + h) * SEQ + ss) * HD + ch] = acc[nt][r];
    }
  }
}

// ---------------------------------------------------------------------------
// Flash-style attention, faithful to the reference's tril-ZEROING semantics:
// masked scores are 0.0 (not -inf) and participate in the full-row softmax.
// One wave owns a 16-query tile of one (b,h); all 8 waves of a block share
// the same (b,h), so the 4KB K/V tiles are staged once per block into LDS
// via double-buffered GLOBAL_LOAD_ASYNC_TO_LDS_B128.
// grid = (16, 32): 16 blocks * 8 waves = 128 q-tiles, y = b*NH + h.
// ---------------------------------------------------------------------------
__global__ __launch_bounds__(256) void attn_kernel(
    const float* __restrict__ qh, const float* __restrict__ kh,
    const float* __restrict__ vh, float* __restrict__ out) {
  __shared__ float k_lds[2][16][KSTRIDE];  // double-buffered K tile
  __shared__ float v_lds[2][16][VSTRIDE];  // double-buffered V tile
  __shared__ float p_s[8][16][PSTRIDE];    // per-wave score/P tile
  __shared__ float m_s[8][16];             // running row max
  __shared__ float l_s[8][16];             // running row sum
  __shared__ float c_s[8][16];             // per-step correction factor

  const int tid  = threadIdx.x;
  const int lane = tid & 31;
  const int wave = tid >> 5;
  const int hi   = lane >> 4;
  const int lo   = lane & 15;

  const int bh = blockIdx.y;
  const int qt = blockIdx.x * 8 + wave;
  const int q0 = qt * 16;

  const float* Q = qh + (size_t)bh * SEQ * HD;
  const float* K = kh + (size_t)bh * SEQ * HD;
  const float* V = vh + (size_t)bh * SEQ * HD;

  // Cooperative async stage of one 16x64 fp32 K tile + V tile (4KB each,
  // contiguous in memory).  Each of the 256 threads moves 16B per tile.
  const int trow = tid >> 4;        // 0..15 : tile row
  const int tseg = (tid & 15) * 4;  // float offset within row, 16B chunks
  const uint32_t k_dst0 =
      (uint32_t)(uintptr_t)(&k_lds[0][0][0]) + (uint32_t)(trow * KSTRIDE + tseg) * 4u;
  const uint32_t k_dst1 =
      (uint32_t)(uintptr_t)(&k_lds[1][0][0]) + (uint32_t)(trow * KSTRIDE + tseg) * 4u;
  const uint32_t v_dst0 =
      (uint32_t)(uintptr_t)(&v_lds[0][0][0]) + (uint32_t)(trow * VSTRIDE + tseg) * 4u;
  const uint32_t v_dst1 =
      (uint32_t)(uintptr_t)(&v_lds[1][0][0]) + (uint32_t)(trow * VSTRIDE + tseg) * 4u;

  auto issue_tiles = [&](int kt_, int buf) {
    const char* kg = (const char*)(K + (size_t)kt_ * 16 * HD) + (size_t)tid * 16;
    const char* vg = (const char*)(V + (size_t)kt_ * 16 * HD) + (size_t)tid * 16;
    async_b128(buf ? k_dst1 : k_dst0, kg);
    async_b128(buf ? v_dst1 : v_dst0, vg);
  };

  // Preload the wave's Q A-fragments for all 16 K-steps across hd=64.
  v2f qa[16];
  {
    const float* qr = Q + (size_t)(q0 + lo) * HD + 2 * hi;
#pragma unroll
    for (int t = 0; t < 16; ++t) qa[t] = *(const v2f*)(qr + t * 4);
  }

  if (lane < 16) {
    m_s[wave][lane] = -INFINITY;
    l_s[wave][lane] = 0.0f;
  }

  v8f o_acc[4] = {v8f_zero(), v8f_zero(), v8f_zero(), v8f_zero()};

  issue_tiles(0, 0);  // prologue

  for (int kt = 0; kt < SEQ / 16; ++kt) {
    const int cur = kt & 1;
    const int nk  = (kt + 1 < SEQ / 16) ? (kt + 1) : (SEQ / 16 - 1);
    issue_tiles(nk, cur ^ 1);  // prefetch next tile into the other buffer

    // Async ops complete in-order per wave: <=2 outstanding means the
    // current tile's 2 loads have landed; barrier publishes all waves' parts.
    wait_asynccnt_le2();
    __syncthreads();

    // ---- S = (Q K^T) * scale for this 16x16 tile (skip if fully masked) ----
    v8f s_acc = v8f_zero();
    if (kt <= qt) {
      const float* kr = &k_lds[cur][lo][2 * hi];
#pragma unroll
      for (int t = 0; t < 16; ++t) {
        v2f b = *(const v2f*)(kr + t * 4);
        s_acc = wmma4(qa[t], b, s_acc);
      }
    }

    // Scale + faithful zero-mask, spill score tile to LDS.
#pragma unroll
    for (int r = 0; r < 8; ++r) {
      const int row = r + 8 * hi;
      float sv = s_acc[r] * SCALE;
      if (kt * 16 + lo > q0 + row) sv = 0.0f;  // tril(): zero, not -inf
      p_s[wave][row][lo] = sv;
    }
    __syncthreads();

    // ---- Online softmax update, one row per lane (lanes 0..15) ----
    if (lane < 16) {
      const int row = lane;
      const float mo = m_s[wave][row];
      float mx = mo;
#pragma unroll
      for (int c = 0; c < 16; ++c) mx = fmaxf(mx, p_s[wave][row][c]);
      const float corr = __expf(mo - mx);  // exp(-inf)=0 on first tile
      float ls = l_s[wave][row] * corr;
#pragma unroll
      for (int c = 0; c < 16; ++c) {
        const float p = __expf(p_s[wave][row][c] - mx);
        p_s[wave][row][c] = p;
        ls += p;
      }
      m_s[wave][row] = mx;
      l_s[wave][row] = ls;
      c_s[wave][row] = corr;
    }
    __syncthreads();

    // ---- Rescale output accumulators by the per-row correction ----
    float cr[8];
#pragma unroll
    for (int r = 0; r < 8; ++r) cr[r] = c_s[wave][r + 8 * hi];
#pragma unroll
    for (int nt = 0; nt < 4; ++nt)
#pragma unroll
      for (int r = 0; r < 8; ++r) o_acc[nt][r] *= cr[r];

    // ---- O += P @ V_tile : A = P (LDS, v2f 8B-aligned), B = V cols (LDS) ---
    const float* vb = &v_lds[cur][0][0];
    const float* pr = &p_s[wave][lo][0];
#pragma unroll
    for (int k0 = 0; k0 < 16; k0 += 4) {
      v2f a = *(const v2f*)(pr + k0 + 2 * hi);
#pragma unroll
      for (int nt = 0; nt < 4; ++nt) {
        v2f b;
        b.x = vb[(k0 + 2 * hi) * VSTRIDE + nt * 16 + lo];
        b.y = vb[(k0 + 2 * hi + 1) * VSTRIDE + nt * 16 + lo];
        o_acc[nt] = wmma4(a, b, o_acc[nt]);
      }
    }
    __syncthreads();  // all reads of buffer `cur` done before it is re-issued
  }

  // ---- Normalize by row sums and store as [b, s, h*hd + ch] ----
  float lr[8];
#pragma unroll
  for (int r = 0; r < 8; ++r) lr[r] = l_s[wave][r + 8 * hi];

  const int b = bh / NH;
  const int h = bh % NH;
#pragma unroll
  for (int r = 0; r < 8; ++r) {
    const int row = q0 + r + 8 * hi;
    const float inv = 1.0f / lr[r];
#pragma unroll
    for (int nt = 0; nt < 4; ++nt) {
      out[((size_t)b * SEQ + row) * DIM + h * HD + nt * 16 + lo] =
          o_acc[nt][r] * inv;
    }
  }
}

extern "C" void kernel_launch(void* const* d_in, const int* in_sizes, int n_in,
                              void* d_out, int out_size, void* d_ws,
                              size_t ws_size, hipStream_t stream) {
  (void)in_sizes; (void)n_in; (void)out_size; (void)ws_size;

  const float* q  = (const float*)d_in[0];
  const float* k  = (const float*)d_in[1];
  const float* v  = (const float*)d_in[2];
  const float* wq = (const float*)d_in[3];
  const float* wk = (const float*)d_in[4];
  const float* wv = (const float*)d_in[5];
  float* out = (float*)d_out;

  float* ws = (float*)d_ws;
  const size_t per = (size_t)NB * SEQ * DIM;  // 4M floats per projection
  float* qh = ws;
  float* kh = ws + per;
  float* vh = ws + 2 * per;

  dim3 pgrid(NB * SEQ / 16 / 8, DIM / 64, 3);  // (32, 16, 3)
  proj_qkv_kernel<<<pgrid, 256, 0, stream>>>(q, k, v, wq, wk, wv, qh, kh, vh);

  dim3 agrid(SEQ / 16 / 8, NB * NH);  // (16, 32)
  attn_kernel<<<agrid, 256, 0, stream>>>(qh, kh, vh, out);
}